// HungarianMatcher_39840116638313
// MI455X (gfx1250) — compile-verified
//
#include <hip/hip_runtime.h>
#include <hip/hip_bf16.h>

// Hungarian matcher (Jonker-Volgenant) for bs=16, T=128 targets, nq=1024 queries.
// One 1024-thread (32 x wave32) workgroup per batch. Cost recomputed on the fly
// from registers+LDS (no cost matrix in memory). JV state in LDS. Wave32
// shuffle argmin with lowest-index tiebreak to match jnp.argmin exactly.

#define NQ 1024
#define NT 128
#define BS 16
#define INFF 1.0e9f

#if defined(__HIP_DEVICE_COMPILE__) && defined(__gfx1250__) && \
    __has_builtin(__builtin_amdgcn_global_load_async_to_lds_b32)
#define USE_ASYNC_LDS 1
#else
#define USE_ASYNC_LDS 0
#endif

__global__ __launch_bounds__(NQ, 1)
void hungarian_jv_kernel(const float* __restrict__ corner_coord,   // [BS, NQ, 2]
                         const float* __restrict__ corner_logits,  // [BS, NQ, 1]
                         const float* __restrict__ tgt_coords,     // [BS, NT, 2]
                         const float* __restrict__ tgt_labels,     // [BS, NT, 1]
                         int* __restrict__ out_rows,               // [BS, NT]
                         int* __restrict__ out_cols)               // [BS, NT]
{
    __shared__ float s_txy[2 * NT];      // interleaved x,y of targets
    __shared__ float s_tp[NT];           // target labels
    __shared__ float s_v[NQ + 1];
    __shared__ float s_minv[NQ + 1];
    __shared__ int   s_way[NQ + 1];
    __shared__ int   s_used[NQ + 1];
    __shared__ int   s_p[NQ + 1];
    __shared__ float s_u[NT + 1];
    __shared__ float s_redv[32];
    __shared__ int   s_redi[32];
    __shared__ float s_delta;
    __shared__ int   s_j0;

    const int b   = blockIdx.x;
    const int tid = (int)threadIdx.x;
    const int j   = tid + 1;             // this thread's column (1..NQ)

    // ---- stage targets into LDS (gfx1250 async global->LDS path if available) ----
#if USE_ASYNC_LDS
    if (tid < 2 * NT) {
        __builtin_amdgcn_global_load_async_to_lds_b32(
            (int*)(tgt_coords + (size_t)b * 2 * NT + tid),
            (int*)&s_txy[tid], 0, 0);
    } else if (tid < 3 * NT) {
        __builtin_amdgcn_global_load_async_to_lds_b32(
            (int*)(tgt_labels + (size_t)b * NT + (tid - 2 * NT)),
            (int*)&s_tp[tid - 2 * NT], 0, 0);
    }
    asm volatile("s_wait_asynccnt 0" ::: "memory");
#else
    if (tid < 2 * NT) {
        s_txy[tid] = tgt_coords[(size_t)b * 2 * NT + tid];
    } else if (tid < 3 * NT) {
        s_tp[tid - 2 * NT] = tgt_labels[(size_t)b * NT + (tid - 2 * NT)];
    }
#endif

    // ---- per-thread query values (live in registers for the whole kernel) ----
    const float qx = corner_coord[(((size_t)b * NQ + tid) << 1) + 0];
    const float qy = corner_coord[(((size_t)b * NQ + tid) << 1) + 1];
    const float ql = corner_logits[(size_t)b * NQ + tid];
    const float qp = 1.0f / (1.0f + __expf(-ql));

    // ---- init persistent JV state ----
    s_v[j] = 0.0f;
    s_p[j] = 0;
    if (tid == 0) { s_v[0] = 0.0f; s_p[0] = 0; }
    if (tid <= NT) s_u[tid] = 0.0f;
    __syncthreads();

    for (int i = 1; i <= NT; ++i) {
        // per-row init
        s_minv[j] = INFF; s_way[j] = 0; s_used[j] = 0;
        if (tid == 0) {
            s_minv[0] = INFF; s_way[0] = 0; s_used[0] = 1;  // j0 starts at 0
            s_p[0] = i; s_j0 = 0;
        }
        __syncthreads();

        while (true) {
            const int j0 = s_j0;
            if (s_p[j0] == 0) break;                 // uniform (LDS after barrier)

            const int   i0  = s_p[j0];
            const float ui0 = s_u[i0];
            const int   t   = i0 - 1;
            // cost(i0-1, j-1) recomputed: 5*(|dx|+|dy|) + |dp|
            const float aij = 5.0f * (fabsf(qx - s_txy[2 * t]) +
                                      fabsf(qy - s_txy[2 * t + 1]))
                              + fabsf(qp - s_tp[t]);
            const float cur = aij - ui0 - s_v[j];
            const int wasused = s_used[j];
            float mv = s_minv[j];
            if (!wasused && cur < mv) { mv = cur; s_minv[j] = cur; s_way[j] = j0; }

            float val = wasused ? INFF : mv;
            int   idx = j;
            // wave32 butterfly argmin (lowest index wins ties, like jnp.argmin)
            #pragma unroll
            for (int off = 16; off; off >>= 1) {
                const float ov = __shfl_xor(val, off, 32);
                const int   oi = __shfl_xor(idx, off, 32);
                if (ov < val || (ov == val && oi < idx)) { val = ov; idx = oi; }
            }
            if ((tid & 31) == 0) { s_redv[tid >> 5] = val; s_redi[tid >> 5] = idx; }
            __syncthreads();

            if (tid < 32) {
                val = s_redv[tid]; idx = s_redi[tid];
                #pragma unroll
                for (int off = 16; off; off >>= 1) {
                    const float ov = __shfl_xor(val, off, 32);
                    const int   oi = __shfl_xor(idx, off, 32);
                    if (ov < val || (ov == val && oi < idx)) { val = ov; idx = oi; }
                }
                if (tid == 0) {
                    s_delta = val;
                    s_used[idx] = 1;   // mark next j0 used (matches body-entry mark)
                    s_j0 = idx;
                }
            }
            __syncthreads();

            const float delta = s_delta;
            // conflict-free scatter: used columns map to distinct rows
            if (wasused) { s_u[s_p[j]] += delta; s_v[j] -= delta; }
            else         { s_minv[j] -= delta; }
            if (tid == 0) { s_u[s_p[0]] += delta; s_v[0] -= delta; }  // column 0
            __syncthreads();
        }

        // augmenting-path flip (short; serial on thread 0)
        if (tid == 0) {
            int j0 = s_j0;
            while (j0 != 0) {
                const int j1 = s_way[j0];
                s_p[j0] = s_p[j1];
                j0 = j1;
            }
        }
        __syncthreads();
    }

    // ordered compaction: rows = ascending matched query ids, cols = p[q+1]-1
    if (tid == 0) {
        int cnt = 0;
        for (int q = 0; q < NQ; ++q) {
            const int pr = s_p[q + 1];
            if (pr > 0) {
                out_rows[b * NT + cnt] = q;
                out_cols[b * NT + cnt] = pr - 1;
                ++cnt;
            }
        }
    }
}

extern "C" void kernel_launch(void* const* d_in, const int* in_sizes, int n_in,
                              void* d_out, int out_size, void* d_ws, size_t ws_size,
                              hipStream_t stream) {
    (void)in_sizes; (void)n_in; (void)d_ws; (void)ws_size; (void)out_size;
    const float* corner_coord  = (const float*)d_in[0];  // [16,1024,2]
    const float* corner_logits = (const float*)d_in[1];  // [16,1024,1]
    const float* tgt_coords    = (const float*)d_in[2];  // [16,128,2]
    const float* tgt_labels    = (const float*)d_in[3];  // [16,128,1]

    int* out_rows = (int*)d_out;            // first BS*NT int32
    int* out_cols = out_rows + BS * NT;     // next  BS*NT int32

    hipLaunchKernelGGL(hungarian_jv_kernel, dim3(BS), dim3(NQ), 0, stream,
                       corner_coord, corner_logits, tgt_coords, tgt_labels,
                       out_rows, out_cols);
}